// VNAttention_49529562857771
// MI455X (gfx1250) — compile-verified
//
#include <hip/hip_runtime.h>
#include <math.h>

// ---------------------------------------------------------------------------
// VN-Attention for MI455X (gfx1250), full-f32 via V_WMMA_F32_16X16X4_F32.
// ~64 GFLOP vs ~50MB HBM traffic -> matrix-pipe bound. All GEMM stages on
// WMMA. Tile staging: TDM tensor_load_to_lds (TENSORcnt) for the attention
// K tile (2D descriptor w/ LDS padding), GLOBAL_LOAD_ASYNC_TO_LDS (ASYNCcnt)
// elsewhere. LDS layouts padded/transposed so every WMMA fragment is one
// aligned ds_load_b64 into an even VGPR pair.
// ---------------------------------------------------------------------------

#define HEADS     8
#define DIM       256
#define DIM_COOR  3
#define DIM_INNER 512
#define NSEQ      2048
#define BATCH     2
#define DC        192                      // DIM_HEAD*DIM_COOR
#define SCALE_F   0.07216878364870323f     // 192^-0.5
#define NEG_BIG   (-3.402823466e38f)

// lds_k row: 192 data DWORDs + 4-DWORD pad every 64 DWORDs -> stride 204
#define KROW      204
#define KOFS(kk)  ((kk) + 4 * ((kk) / 64))

typedef float v2f __attribute__((ext_vector_type(2)));
typedef float v8f __attribute__((ext_vector_type(8)));
typedef int   vi4 __attribute__((vector_size(4 * sizeof(int))));

#if defined(__gfx1250__) && __has_builtin(__builtin_amdgcn_global_load_async_to_lds_b128) && \
    __has_builtin(__builtin_amdgcn_global_load_async_to_lds_b32) && \
    __has_builtin(__builtin_amdgcn_s_wait_asynccnt)
#define USE_ASYNC 1
#else
#define USE_ASYNC 0
#endif

#if defined(__gfx1250__) && __has_builtin(__builtin_amdgcn_tensor_load_to_lds) && \
    __has_builtin(__builtin_amdgcn_s_wait_tensorcnt)
#define USE_TDM 1
#else
#define USE_TDM 0
#endif

// async builtin signatures (from hipcc diagnostics):
//   b128: (AS1 vi4* src, AS3 vi4* dst, imm offset, imm cpol)
//   b32 : (AS1 int* src, AS3 int* dst, imm offset, imm cpol)
#define GP128(p) ((__attribute__((address_space(1))) vi4*)(uintptr_t)(p))
#define LP128(p) ((__attribute__((address_space(3))) vi4*)(unsigned)(uintptr_t)(p))
#define GP32(p)  ((__attribute__((address_space(1))) int*)(uintptr_t)(p))
#define LP32(p)  ((__attribute__((address_space(3))) int*)(unsigned)(uintptr_t)(p))

__device__ __forceinline__ void async_cp_b128(const void* g, void* l) {
#if USE_ASYNC
  __builtin_amdgcn_global_load_async_to_lds_b128(GP128(g), LP128(l), 0, 0);
#else
  *(float4*)l = *(const float4*)g;
#endif
}
__device__ __forceinline__ void async_cp_b32(const void* g, void* l) {
#if USE_ASYNC
  __builtin_amdgcn_global_load_async_to_lds_b32(GP32(g), LP32(l), 0, 0);
#else
  *(float*)l = *(const float*)g;
#endif
}
__device__ __forceinline__ void async_wait() {
#if USE_ASYNC
  __builtin_amdgcn_s_wait_asynccnt(0);
#endif
}

#if USE_TDM
typedef unsigned int u32x4 __attribute__((ext_vector_type(4)));
typedef int          i32x8 __attribute__((ext_vector_type(8)));
typedef int          i32x4 __attribute__((ext_vector_type(4)));

// 2D TDM load: 32 x 192 f32 tile, global row stride 192, into LDS at ldsoff
// with 4-DWORD padding every 64 DWORDs (D# group0/1 per CDNA5 ISA 8.3/8.4).
// This toolchain exposes the 6-arg builtin form: (g0, g1, g2, g3, x8, cpol).
__device__ __forceinline__ void tdm_load_k(const float* gsrc, unsigned ldsoff) {
  uint64_t ga = (uint64_t)(uintptr_t)gsrc;
  u32x4 g0 = { 1u,                                   // count=1, user mode
               ldsoff,                               // lds_addr (bytes)
               (unsigned)ga,                         // global_addr[31:0]
               ((unsigned)(ga >> 32) & 0x01FFFFFFu) | (2u << 30) }; // +type=2
  i32x8 g1 = { (int)((2u << 16)      // data_size = 4B
                   | (1u << 20)      // pad_enable
                   | (5u << 22)      // pad_interval: 64 DWORDs
                   | (3u << 25)),    // pad_amount : 4 DWORDs
               (int)(192u << 16),    // tensor_dim0 = 192 (bits 79:48, low)
               (int)(2048u << 16),   // tensor_dim1 = 2048 (bits 111:80, low)
               (int)(192u << 16),    // tile_dim0 = 192 (bits 127:112)
               32,                   // tile_dim1 = 32 (bits 143:128)
               192,                  // tensor_dim0_stride = 192 (bits 207:160)
               0, 0 };
  i32x4 gz4 = { 0, 0, 0, 0 };
  i32x8 gz8 = { 0, 0, 0, 0, 0, 0, 0, 0 };
  __builtin_amdgcn_tensor_load_to_lds(g0, g1, gz4, gz4, gz8, 0);
}
__device__ __forceinline__ void tdm_wait() {
  __builtin_amdgcn_s_wait_tensorcnt(0);
}
#endif

// D = A(16x4)*B(4x16)+C (f32). frag: 2 floats/lane, k = 2*(lane/16)+comp;
// C/D: row m = r + 8*(lane/16), col n = lane%16.
__device__ __forceinline__ v8f wmma4(v2f a, v2f b, v8f c) {
  return __builtin_amdgcn_wmma_f32_16x16x4_f32(false, a, false, b,
                                               (short)0, c, false, false);
}

// ---------------------------------------------------------------------------
// WMMA GEMM:  C[o,n] = sum_k W[o,k] * X[b][n][k*3 + c]   per (b,c)
// grid: (NSEQ/64, M/128, BATCH*3), block 128 (4 waves, 32 M-rows each).
// mode 0: head layout dst[(b*8+head)][n][d*3+c], rows >= 512 -> dstB.
// mode 1: dstA[((b*NSEQ+n)*DIM + o)*3 + c]  (final projection -> d_out).
// ---------------------------------------------------------------------------
__global__ __launch_bounds__(128)
void proj_kernel(const float* __restrict__ W, const float* __restrict__ X,
                 float* __restrict__ dstA, float* __restrict__ dstB,
                 int M, int K, int in_row_stride, int mode)
{
  __shared__ float lds_a [128][20];   // W slab, row-major, pad 20 (16B rows)
  __shared__ float lds_bt[64][20];    // X slab TRANSPOSED [n][k], pad 20

  const int n0   = blockIdx.x * 64;
  const int m0   = blockIdx.y * 128;
  const int b    = blockIdx.z / 3;
  const int c    = blockIdx.z % 3;
  const int tid  = threadIdx.x;
  const int lane = tid & 31;
  const int w    = tid >> 5;
  const int hi   = lane >> 4;
  const int ln   = lane & 15;

  const float* Xb = X + (size_t)b * NSEQ * in_row_stride;

  v8f acc[2][4];
  #pragma unroll
  for (int ms = 0; ms < 2; ++ms)
    #pragma unroll
    for (int t = 0; t < 4; ++t) acc[ms][t] = (v8f){0,0,0,0,0,0,0,0};

  for (int k0 = 0; k0 < K; k0 += 16) {
    __syncthreads();                       // previous chunk fully consumed
    {   // stage A: 128x16 of W (async b128, 4 per thread)
      const float* src = W + (size_t)(m0 + tid) * K + k0;
      async_cp_b128(src,      &lds_a[tid][0]);
      async_cp_b128(src + 4,  &lds_a[tid][4]);
      async_cp_b128(src + 8,  &lds_a[tid][8]);
      async_cp_b128(src + 12, &lds_a[tid][12]);
    }
    {   // stage B transposed: lds_bt[n][k] = X[b][n0+n][(k0+k)*3+c]
      int n_l = tid >> 1;
      int kh  = (tid & 1) * 8;
      const float* src = Xb + (size_t)(n0 + n_l) * in_row_stride
                            + (size_t)(k0 + kh) * 3 + c;
      #pragma unroll
      for (int u = 0; u < 8; ++u)
        async_cp_b32(src + (size_t)u * 3, &lds_bt[n_l][kh + u]);
    }
    async_wait();
    __syncthreads();

    #pragma unroll
    for (int kc = 0; kc < 4; ++kc) {
      int kk = 4 * kc + 2 * hi;
      v2f a0 = *(const v2f*)&lds_a[32 * w + ln][kk];
      v2f a1 = *(const v2f*)&lds_a[32 * w + 16 + ln][kk];
      #pragma unroll
      for (int t = 0; t < 4; ++t) {
        v2f bb = *(const v2f*)&lds_bt[16 * t + ln][kk];
        acc[0][t] = wmma4(a0, bb, acc[0][t]);
        acc[1][t] = wmma4(a1, bb, acc[1][t]);
      }
    }
  }

  #pragma unroll
  for (int ms = 0; ms < 2; ++ms) {
    #pragma unroll
    for (int t = 0; t < 4; ++t) {
      #pragma unroll
      for (int r = 0; r < 8; ++r) {
        int o = m0 + 32 * w + 16 * ms + r + 8 * hi;
        int n = n0 + 16 * t + ln;
        float val = acc[ms][t][r];
        if (mode == 0) {
          float* T; int oo;
          if (o < DIM_INNER) { T = dstA; oo = o; } else { T = dstB; oo = o - DIM_INNER; }
          int head = oo >> 6, d = oo & 63;
          T[((size_t)(b * HEADS + head) * NSEQ + n) * DC + d * 3 + c] = val;
        } else {
          dstA[(((size_t)b * NSEQ + n) * DIM + o) * 3 + c] = val;
        }
      }
    }
  }
}

// ---------------------------------------------------------------------------
__global__ __launch_bounds__(256)
void sq_kernel(const float* __restrict__ Q, const float* __restrict__ Km,
               float* __restrict__ sqq, float* __restrict__ sqk)
{
  int idx  = blockIdx.x * blockDim.x + threadIdx.x;
  int half = BATCH * HEADS * NSEQ;
  const float* src; float* dst; int row;
  if (idx < half) { src = Q;  dst = sqq; row = idx; }
  else            { src = Km; dst = sqk; row = idx - half; }
  const float* p = src + (size_t)row * DC;
  float s = 0.f;
  #pragma unroll
  for (int i = 0; i < DC; i += 4) {
    float4 v = *(const float4*)(p + i);
    s += v.x * v.x + v.y * v.y + v.z * v.z + v.w * v.w;
  }
  dst[row] = s;
}

// ---------------------------------------------------------------------------
// Flash attention. grid: (NSEQ/64, BATCH*HEADS), block 128 (4 waves x 16 i).
// K tile via TDM (wave 0 issues, TENSORcnt); V tile transposed via async b32.
// logits = (2*QK - |q|^2 - |k|^2)*SCALE, online softmax, O += P*V.
// ---------------------------------------------------------------------------
__global__ __launch_bounds__(128)
void attn_kernel(const float* __restrict__ Q, const float* __restrict__ Km,
                 const float* __restrict__ V, const float* __restrict__ sqq,
                 const float* __restrict__ sqk,
                 const unsigned char* __restrict__ mask,
                 float* __restrict__ O)
{
  __shared__ float lds_k [32][KROW];    // K tile, TDM-padded rows (204 dw)
  __shared__ float lds_vt[192][36];     // V tile TRANSPOSED [dc][j], pad 36
  __shared__ float lds_p [4][16][34];   // per-wave P transpose tile

  const int bh   = blockIdx.y;
  const int b    = bh >> 3;
  const int h    = bh & 7;
  const int i0   = blockIdx.x * 64;
  const int tid  = threadIdx.x;
  const int lane = tid & 31;
  const int w    = tid >> 5;
  const int hi   = lane >> 4;
  const int ln   = lane & 15;
  const int iw   = i0 + 16 * w;

  const float* Qb = Q  + (size_t)bh * NSEQ * DC;
  const float* Kb = Km + (size_t)bh * NSEQ * DC;
  const float* Vb = V  + (size_t)bh * NSEQ * DC;
  const unsigned char* mb = mask + (size_t)b * NSEQ;

  // Q A-fragments (16x192) resident in registers: 48 x v2f (b64 loads)
  v2f qa[48];
  {
    const float* qrow = Qb + (size_t)(iw + ln) * DC;
    #pragma unroll
    for (int kc = 0; kc < 48; ++kc)
      qa[kc] = *(const v2f*)(qrow + 4 * kc + 2 * hi);
  }
  float qs[8];
  #pragma unroll
  for (int r = 0; r < 8; ++r)
    qs[r] = sqq[(size_t)bh * NSEQ + iw + r + 8 * hi];

  v8f oacc[12];
  #pragma unroll
  for (int t = 0; t < 12; ++t) oacc[t] = (v8f){0,0,0,0,0,0,0,0};
  float rowm[8], rowl[8];
  #pragma unroll
  for (int r = 0; r < 8; ++r) { rowm[r] = -INFINITY; rowl[r] = 0.f; }

  for (int j0 = 0; j0 < NSEQ; j0 += 32) {
    __syncthreads();
#if USE_TDM
    if (w == 0)       // one DMA per workgroup; TDM ignores EXEC, per-wave issue
      tdm_load_k(Kb + (size_t)j0 * DC, (unsigned)(uintptr_t)&lds_k[0][0]);
#else
    {   // K tile 32x192 into padded rows, coalesced async b128
      #pragma unroll
      for (int u = 0; u < 12; ++u) {
        int f   = u * 128 + tid;          // 0..1535 float4 chunks
        int jr  = f / 48;
        int dcb = (f % 48) * 4;
        async_cp_b128(Kb + (size_t)(j0 + jr) * DC + dcb, &lds_k[jr][KOFS(dcb)]);
      }
    }
#endif
    {   // V tile transposed: lds_vt[dc][j] (async b32 scatter)
      #pragma unroll
      for (int u = 0; u < 12; ++u) {
        int g  = u * 128 + tid;           // 0..1535 (dc, j-quad) groups
        int jb = g / 192;                 // 0..7 -> j quad
        int dc = g % 192;
        const float* src = Vb + (size_t)(j0 + 4 * jb) * DC + dc;
        #pragma unroll
        for (int e = 0; e < 4; ++e)
          async_cp_b32(src + (size_t)e * DC, &lds_vt[dc][4 * jb + e]);
      }
    }
    async_wait();
#if USE_TDM
    if (w == 0) tdm_wait();
#endif
    __syncthreads();

    // S(16x32) = Q * K^T : 2x48 WMMA k-steps, b64 fragments
    v8f s0 = (v8f){0,0,0,0,0,0,0,0};
    v8f s1 = (v8f){0,0,0,0,0,0,0,0};
    #pragma unroll
    for (int kc = 0; kc < 48; ++kc) {
      int kk  = 4 * kc + 2 * hi;
      int kkp = KOFS(kk);
      v2f b0 = *(const v2f*)&lds_k[ln][kkp];
      v2f b1 = *(const v2f*)&lds_k[16 + ln][kkp];
      s0 = wmma4(qa[kc], b0, s0);
      s1 = wmma4(qa[kc], b1, s1);
    }

    float kq0 = sqk[(size_t)bh * NSEQ + j0 + ln];
    float kq1 = sqk[(size_t)bh * NSEQ + j0 + 16 + ln];
    bool  mk0 = mb[j0 + ln] != 0;
    bool  mk1 = mb[j0 + 16 + ln] != 0;

    // online softmax (row state replicated across each 16-lane group)
    #pragma unroll
    for (int r = 0; r < 8; ++r) {
      float l0 = mk0 ? (2.f * s0[r] - qs[r] - kq0) * SCALE_F : NEG_BIG * SCALE_F;
      float l1 = mk1 ? (2.f * s1[r] - qs[r] - kq1) * SCALE_F : NEG_BIG * SCALE_F;
      float tmax = fmaxf(l0, l1);
      #pragma unroll
      for (int d = 1; d < 16; d <<= 1) tmax = fmaxf(tmax, __shfl_xor(tmax, d, 32));
      float mnew  = fmaxf(rowm[r], tmax);
      float scale = __expf(rowm[r] - mnew);
      float p0 = __expf(l0 - mnew);
      float p1 = __expf(l1 - mnew);
      float tsum = p0 + p1;
      #pragma unroll
      for (int d = 1; d < 16; d <<= 1) tsum += __shfl_xor(tsum, d, 32);
      rowl[r] = rowl[r] * scale + tsum;
      rowm[r] = mnew;
      #pragma unroll
      for (int t = 0; t < 12; ++t) oacc[t][r] *= scale;
      int m = r + 8 * hi;
      lds_p[w][m][ln]      = p0;
      lds_p[w][m][16 + ln] = p1;
    }

    // O(16x192) += P(16x32) * V(32x192): all-b64 fragments
    v2f pa[8];
    #pragma unroll
    for (int kc = 0; kc < 8; ++kc)
      pa[kc] = *(const v2f*)&lds_p[w][ln][4 * kc + 2 * hi];
    #pragma unroll
    for (int t = 0; t < 12; ++t) {
      #pragma unroll
      for (int kc = 0; kc < 8; ++kc) {
        int kk = 4 * kc + 2 * hi;
        v2f bv = *(const v2f*)&lds_vt[16 * t + ln][kk];
        oacc[t] = wmma4(pa[kc], bv, oacc[t]);
      }
    }
  }

  #pragma unroll
  for (int r = 0; r < 8; ++r) {
    int i = iw + r + 8 * hi;
    float inv = 1.f / rowl[r];
    float* orow = O + ((size_t)(b * NSEQ + i)) * (HEADS * DC) + h * DC;
    #pragma unroll
    for (int t = 0; t < 12; ++t) orow[16 * t + ln] = oacc[t][r] * inv;
  }
}

// ---------------------------------------------------------------------------
extern "C" void kernel_launch(void* const* d_in, const int* in_sizes, int n_in,
                              void* d_out, int out_size, void* d_ws, size_t ws_size,
                              hipStream_t stream)
{
  const float*         x     = (const float*)d_in[0];
  const unsigned char* mask  = (const unsigned char*)d_in[1];
  const float*         w_q   = (const float*)d_in[2];
  const float*         w_kv  = (const float*)d_in[3];
  const float*         w_out = (const float*)d_in[4];
  float*               out   = (float*)d_out;

  float* ws  = (float*)d_ws;
  const size_t per = (size_t)BATCH * HEADS * NSEQ * DC;   // 6,291,456 floats
  float* Qw  = ws;
  float* Kw  = Qw + per;
  float* Vw  = Kw + per;
  float* Ow  = Vw + per;                                   // [b][n][512*3]
  float* sqq = Ow + per;
  float* sqk = sqq + (size_t)BATCH * HEADS * NSEQ;

  dim3 blk(128);
  proj_kernel<<<dim3(NSEQ / 64, DIM_INNER / 128, BATCH * 3), blk, 0, stream>>>(
      w_q, x, Qw, nullptr, DIM_INNER, DIM, DIM * DIM_COOR, 0);
  proj_kernel<<<dim3(NSEQ / 64, (2 * DIM_INNER) / 128, BATCH * 3), blk, 0, stream>>>(
      w_kv, x, Kw, Vw, 2 * DIM_INNER, DIM, DIM * DIM_COOR, 0);
  sq_kernel<<<(2 * BATCH * HEADS * NSEQ) / 256, 256, 0, stream>>>(Qw, Kw, sqq, sqk);
  attn_kernel<<<dim3(NSEQ / 64, BATCH * HEADS), blk, 0, stream>>>(
      Qw, Kw, Vw, sqq, sqk, mask, Ow);
  proj_kernel<<<dim3(NSEQ / 64, DIM / 128, BATCH * 3), blk, 0, stream>>>(
      w_out, Ow, out, nullptr, DIM, DIM_INNER, DIM_INNER * DIM_COOR, 1);
}